// Model_89043261980806
// MI455X (gfx1250) — compile-verified
//
#include <hip/hip_runtime.h>
#include <hip/hip_bf16.h>

typedef float v2f __attribute__((ext_vector_type(2)));
typedef float v8f __attribute__((ext_vector_type(8)));

#define NEG_SLOPE 0.01f

// ---------------------------------------------------------------------------
// fp32 WMMA GEMM: C[M,Nc] = A[M,K] @ B[K,Nc], row-major, K <= 128, K % 4 == 0.
// One wave per 16x16 C tile, 4 waves/block. B column-slice is staged in LDS
// zero-padded, so the inner loop is branch-free:
//   global_load_b64 (A pair) + 2x ds_load_b32 (B pair) + v_wmma_f32_16x16x4_f32
// M is assumed a multiple of 16 at tile granularity (true here: 50000 / 1.6M);
// A row index is clamped so duplicate rows are harmless and never stored.
// ---------------------------------------------------------------------------
__global__ void gemm_f32_wmma(const float* __restrict__ A,
                              const float* __restrict__ B,
                              float* __restrict__ C,
                              int M, int Nc, int K) {
    __shared__ float sB[128 * 16];                 // K <= 128
    const int tileN = blockIdx.y * 16;

    // cooperative zero-padded stage of B[:, tileN:tileN+16]
    for (int idx = threadIdx.x; idx < K * 16; idx += blockDim.x) {
        const int k = idx >> 4, c = idx & 15;
        const int col = tileN + c;
        sB[idx] = (col < Nc) ? B[(size_t)k * Nc + col] : 0.0f;
    }
    __syncthreads();

    const int wave = threadIdx.x >> 5;
    const int lane = threadIdx.x & 31;
    const int half = lane >> 4;                    // K-pair / M-high-half select
    const int r    = lane & 15;
    const int tileM = (blockIdx.x * 4 + wave) * 16;
    if (tileM >= M) return;

    const int rowA = min(tileM + r, M - 1);        // clamp -> unconditional loads
    const float* aPtr = A + (size_t)rowA * K + 2 * half;  // 8B aligned
    const float* bPtr = sB + (2 * half) * 16 + r;

    v8f acc = {};
#pragma unroll 4
    for (int k = 0; k < K; k += 4) {
        v2f a = *(const v2f*)(aPtr + k);           // A: K=k..k+1 (lo) / k+2..k+3 (hi)
        v2f b;
        b.x = bPtr[k * 16];                        // B row k (+2 for hi half)
        b.y = bPtr[(k + 1) * 16];
        acc = __builtin_amdgcn_wmma_f32_16x16x4_f32(
            /*neg_a=*/false, a, /*neg_b=*/false, b,
            /*c_mod=*/(short)0, acc, /*reuse_a=*/false, /*reuse_b=*/false);
    }

    const int colB = tileN + r;
    if (colB < Nc) {
#pragma unroll
        for (int i = 0; i < 8; ++i) {
            const int row = tileM + i + 8 * half;  // C: VGPR i -> M = i (+8 hi half)
            if (row < M) C[(size_t)row * Nc + colB] = acc[i];
        }
    }
}

// ---------------------------------------------------------------------------
// Zero-fill (accumulators; 0u is also the ordered-max encoding floor)
// ---------------------------------------------------------------------------
__global__ void zero_f32(float* __restrict__ p, long long n) {
    long long i = (long long)blockIdx.x * blockDim.x + threadIdx.x;
    if (i < n) p[i] = 0.0f;
}

// ordered-uint encoding so atomicMax(uint) == float max
__device__ __forceinline__ unsigned int f32_ord(float f) {
    unsigned int u = __float_as_uint(f);
    return (u & 0x80000000u) ? ~u : (u | 0x80000000u);
}
__device__ __forceinline__ float ord_f32(unsigned int u) {
    u = (u & 0x80000000u) ? (u & 0x7fffffffu) : ~u;
    return __uint_as_float(u);
}

// ---------------------------------------------------------------------------
// Layer-1 fused edge kernel: f = leaky(Xi[src] + Xj[dst] + ef@Wfij + b),
// per-head logits, head-mean edge features. All streams via float4 (b128).
// ---------------------------------------------------------------------------
__global__ void edge1_fused(const float* __restrict__ ef,
                            const int* __restrict__ src,
                            const int* __restrict__ dst,
                            const float* __restrict__ Xi,
                            const float* __restrict__ Xj,
                            const float* __restrict__ Wf,    // [16,48]
                            const float* __restrict__ b,     // [48]
                            const float* __restrict__ attn,  // [3,16]
                            float* __restrict__ elog,        // [E,3]
                            float* __restrict__ fe_mean,     // [E,16]
                            int E) {
    __shared__ float sW[16 * 48];
    __shared__ float sb[48];
    __shared__ float sat[48];
    for (int i = threadIdx.x; i < 16 * 48; i += blockDim.x) sW[i] = Wf[i];
    if (threadIdx.x < 48) { sb[threadIdx.x] = b[threadIdx.x]; sat[threadIdx.x] = attn[threadIdx.x]; }
    __syncthreads();

    const int e = blockIdx.x * blockDim.x + threadIdx.x;
    if (e >= E) return;
    const int s = src[e], d = dst[e];

    float f[48];
    const float4* xi4 = (const float4*)(Xi + (size_t)s * 48);   // 192B rows, 16B aligned
    const float4* xj4 = (const float4*)(Xj + (size_t)d * 48);
#pragma unroll
    for (int q = 0; q < 12; ++q) {
        const float4 a = xi4[q];
        const float4 c = xj4[q];
        f[4 * q + 0] = sb[4 * q + 0] + a.x + c.x;
        f[4 * q + 1] = sb[4 * q + 1] + a.y + c.y;
        f[4 * q + 2] = sb[4 * q + 2] + a.z + c.z;
        f[4 * q + 3] = sb[4 * q + 3] + a.w + c.w;
    }

    float ev[16];
    const float4* ep4 = (const float4*)(ef + (size_t)e * 16);   // 64B rows
#pragma unroll
    for (int q = 0; q < 4; ++q) {
        const float4 v = ep4[q];
        ev[4 * q + 0] = v.x; ev[4 * q + 1] = v.y; ev[4 * q + 2] = v.z; ev[4 * q + 3] = v.w;
    }
#pragma unroll
    for (int k = 0; k < 16; ++k) {
        const float w = ev[k];
#pragma unroll
        for (int j = 0; j < 48; ++j) f[j] += w * sW[k * 48 + j];
    }
#pragma unroll
    for (int j = 0; j < 48; ++j) { float x = f[j]; f[j] = x > 0.0f ? x : NEG_SLOPE * x; }

#pragma unroll
    for (int h = 0; h < 3; ++h) {
        float acc = 0.0f;
#pragma unroll
        for (int t = 0; t < 16; ++t) acc += f[h * 16 + t] * sat[h * 16 + t];
        elog[(size_t)e * 3 + h] = acc;
    }

    float4* fm4 = (float4*)(fe_mean + (size_t)e * 16);
#pragma unroll
    for (int q = 0; q < 4; ++q) {
        float4 o;
        o.x = (f[4 * q + 0] + f[16 + 4 * q + 0] + f[32 + 4 * q + 0]) * (1.0f / 3.0f);
        o.y = (f[4 * q + 1] + f[16 + 4 * q + 1] + f[32 + 4 * q + 1]) * (1.0f / 3.0f);
        o.z = (f[4 * q + 2] + f[16 + 4 * q + 2] + f[32 + 4 * q + 2]) * (1.0f / 3.0f);
        o.w = (f[4 * q + 3] + f[16 + 4 * q + 3] + f[32 + 4 * q + 3]) * (1.0f / 3.0f);
        fm4[q] = o;
    }
}

// ---------------------------------------------------------------------------
// Segment softmax: pass 1 max, pass 2 exp + sum (generic in head count H)
// ---------------------------------------------------------------------------
__global__ void seg_max(const float* __restrict__ elog, const int* __restrict__ dst,
                        unsigned int* __restrict__ m, int E, int H) {
    const long long t = (long long)blockIdx.x * blockDim.x + threadIdx.x;
    if (t >= (long long)E * H) return;
    const int e = (int)(t / H), h = (int)(t - (long long)e * H);
    atomicMax(m + (size_t)dst[e] * H + h, f32_ord(elog[t]));
}

__global__ void seg_expsum(const float* __restrict__ elog, const int* __restrict__ dst,
                           const unsigned int* __restrict__ m,
                           float* __restrict__ s, float* __restrict__ ex, int E, int H) {
    const long long t = (long long)blockIdx.x * blockDim.x + threadIdx.x;
    if (t >= (long long)E * H) return;
    const int e = (int)(t / H), h = (int)(t - (long long)e * H);
    const float mv = ord_f32(m[(size_t)dst[e] * H + h]);
    const float evv = __expf(elog[t] - mv);
    ex[t] = evv;
    atomicAdd(s + (size_t)dst[e] * H + h, evv);
}

// ---------------------------------------------------------------------------
// Layer-1 aggregation fused with head-mean: one wave per edge, lane = column.
// hmean[dst, j] += (1/3) * sum_h a[e,h] * Hn1[src, h*32 + j]
// ---------------------------------------------------------------------------
__global__ void agg1(const int* __restrict__ src, const int* __restrict__ dst,
                     const float* __restrict__ Hn1,  // [N, 96]
                     const float* __restrict__ ex,   // [E, 3]
                     const float* __restrict__ ssum, // [N, 3]
                     float* __restrict__ hmean,      // [N, 32]
                     int E) {
    const long long gid = (long long)blockIdx.x * blockDim.x + threadIdx.x;
    const long long e = gid >> 5;
    const int lane = threadIdx.x & 31;
    if (e >= E) return;
    const int s = src[e], d = dst[e];
    const float a0 = ex[e * 3 + 0] / ssum[(size_t)d * 3 + 0];
    const float a1 = ex[e * 3 + 1] / ssum[(size_t)d * 3 + 1];
    const float a2 = ex[e * 3 + 2] / ssum[(size_t)d * 3 + 2];
    const float* hp = Hn1 + (size_t)s * 96;
    const float val = (a0 * hp[lane] + a1 * hp[32 + lane] + a2 * hp[64 + lane]) * (1.0f / 3.0f);
    atomicAdd(hmean + (size_t)d * 32 + lane, val);
}

// ---------------------------------------------------------------------------
// Layer-2 fused edge kernel (1 head, 8 cols)
// ---------------------------------------------------------------------------
__global__ void edge2_fused(const float* __restrict__ fe,  // [E,16]
                            const int* __restrict__ src,
                            const int* __restrict__ dst,
                            const float* __restrict__ Xi,  // [N,8]
                            const float* __restrict__ Xj,  // [N,8]
                            const float* __restrict__ Wf,  // [16,8]
                            const float* __restrict__ b,   // [8]
                            const float* __restrict__ attn,// [8]
                            float* __restrict__ elog,      // [E]
                            int E) {
    __shared__ float sW[16 * 8];
    __shared__ float sb[8];
    __shared__ float sat[8];
    for (int i = threadIdx.x; i < 16 * 8; i += blockDim.x) sW[i] = Wf[i];
    if (threadIdx.x < 8) { sb[threadIdx.x] = b[threadIdx.x]; sat[threadIdx.x] = attn[threadIdx.x]; }
    __syncthreads();

    const int e = blockIdx.x * blockDim.x + threadIdx.x;
    if (e >= E) return;
    const int s = src[e], d = dst[e];

    float f[8];
    const float4* xi4 = (const float4*)(Xi + (size_t)s * 8);  // 32B rows
    const float4* xj4 = (const float4*)(Xj + (size_t)d * 8);
#pragma unroll
    for (int q = 0; q < 2; ++q) {
        const float4 a = xi4[q];
        const float4 c = xj4[q];
        f[4 * q + 0] = sb[4 * q + 0] + a.x + c.x;
        f[4 * q + 1] = sb[4 * q + 1] + a.y + c.y;
        f[4 * q + 2] = sb[4 * q + 2] + a.z + c.z;
        f[4 * q + 3] = sb[4 * q + 3] + a.w + c.w;
    }
    const float4* fp4 = (const float4*)(fe + (size_t)e * 16);
#pragma unroll
    for (int q = 0; q < 4; ++q) {
        const float4 v = fp4[q];
        const float evv[4] = { v.x, v.y, v.z, v.w };
#pragma unroll
        for (int kk = 0; kk < 4; ++kk) {
            const int k = 4 * q + kk;
#pragma unroll
            for (int j = 0; j < 8; ++j) f[j] += evv[kk] * sW[k * 8 + j];
        }
    }
    float acc = 0.0f;
#pragma unroll
    for (int j = 0; j < 8; ++j) {
        float x = f[j]; x = x > 0.0f ? x : NEG_SLOPE * x;
        acc += x * sat[j];
    }
    elog[e] = acc;
}

// ---------------------------------------------------------------------------
// Layer-2 aggregation: h2[dst, 0..4] += a * Hn2[src, 0..4]
// ---------------------------------------------------------------------------
__global__ void agg2(const int* __restrict__ src, const int* __restrict__ dst,
                     const float* __restrict__ Hn2,  // [N,5]
                     const float* __restrict__ ex,   // [E]
                     const float* __restrict__ ssum, // [N]
                     float* __restrict__ h2,         // [N,5]
                     int E) {
    const int e = blockIdx.x * blockDim.x + threadIdx.x;
    if (e >= E) return;
    const int s = src[e], d = dst[e];
    const float a = ex[e] / ssum[d];
    const float* hp = Hn2 + (size_t)s * 5;
#pragma unroll
    for (int j = 0; j < 5; ++j) atomicAdd(h2 + (size_t)d * 5 + j, a * hp[j]);
}

// ---------------------------------------------------------------------------
// Predictor: score[e] = [h2[src], h2[dst]] . Wp + bp
// ---------------------------------------------------------------------------
__global__ void predict(const int* __restrict__ src, const int* __restrict__ dst,
                        const float* __restrict__ h2, const float* __restrict__ Wp,
                        const float* __restrict__ bp, float* __restrict__ out, int E) {
    const int e = blockIdx.x * blockDim.x + threadIdx.x;
    if (e >= E) return;
    const int s = src[e], d = dst[e];
    float acc = bp[0];
    const float* hs = h2 + (size_t)s * 5;
    const float* hd = h2 + (size_t)d * 5;
#pragma unroll
    for (int j = 0; j < 5; ++j) acc += hs[j] * Wp[j];
#pragma unroll
    for (int j = 0; j < 5; ++j) acc += hd[j] * Wp[5 + j];
    out[e] = acc;
}

// ---------------------------------------------------------------------------
extern "C" void kernel_launch(void* const* d_in, const int* in_sizes, int n_in,
                              void* d_out, int out_size, void* d_ws, size_t ws_size,
                              hipStream_t stream) {
    const float* nf     = (const float*)d_in[0];
    const float* ef     = (const float*)d_in[1];
    const int*   src    = (const int*)d_in[2];
    const int*   dst    = (const int*)d_in[3];
    const float* Wni1   = (const float*)d_in[4];
    const float* Wnj1   = (const float*)d_in[5];
    const float* Wfij1  = (const float*)d_in[6];
    const float* b1     = (const float*)d_in[7];
    const float* attn1  = (const float*)d_in[8];
    const float* Wnode1 = (const float*)d_in[9];
    const float* Wni2   = (const float*)d_in[10];
    const float* Wnj2   = (const float*)d_in[11];
    const float* Wfij2  = (const float*)d_in[12];
    const float* b2     = (const float*)d_in[13];
    const float* attn2  = (const float*)d_in[14];
    const float* Wnode2 = (const float*)d_in[15];
    const float* Wp     = (const float*)d_in[16];
    const float* bp     = (const float*)d_in[17];

    const int N = in_sizes[0] / 128;
    const int E = in_sizes[2];

    // workspace carve-up (all segments 16B-aligned: N, E multiples of 16)
    float* p = (float*)d_ws;
    float* Xi1   = p; p += (size_t)N * 48;
    float* Xj1   = p; p += (size_t)N * 48;
    float* Hn1   = p; p += (size_t)N * 96;
    float* elog1 = p; p += (size_t)E * 3;
    float* ex1   = p; p += (size_t)E * 3;
    float* fe_m  = p; p += (size_t)E * 16;
    float* Xi2   = p; p += (size_t)N * 8;
    float* Xj2   = p; p += (size_t)N * 8;
    float* Hn2   = p; p += (size_t)N * 5;
    float* elog2 = p; p += (size_t)E;
    float* ex2   = p; p += (size_t)E;
    // accumulator zone (zeroed each call)
    float* accBase = p;
    float* m1    = p; p += (size_t)N * 3;   // ordered-uint max
    float* s1    = p; p += (size_t)N * 3;
    float* hmean = p; p += (size_t)N * 32;
    float* m2    = p; p += (size_t)N;       // ordered-uint max
    float* s2    = p; p += (size_t)N;
    float* h2    = p; p += (size_t)N * 5;
    const long long accCount = (long long)N * 45;

    const int T = 256;
    zero_f32<<<(unsigned)((accCount + T - 1) / T), T, 0, stream>>>(accBase, accCount);

    // ---- layer 1 node projections (WMMA fp32) ----
    dim3 gblk(128);  // 4 waves per block
    dim3 g1((N + 63) / 64, 3);
    gemm_f32_wmma<<<g1, gblk, 0, stream>>>(nf, Wni1, Xi1, N, 48, 128);
    gemm_f32_wmma<<<g1, gblk, 0, stream>>>(nf, Wnj1, Xj1, N, 48, 128);
    dim3 g1n((N + 63) / 64, 6);
    gemm_f32_wmma<<<g1n, gblk, 0, stream>>>(nf, Wnode1, Hn1, N, 96, 128);

    // ---- layer 1 edge pipeline ----
    edge1_fused<<<(E + T - 1) / T, T, 0, stream>>>(ef, src, dst, Xi1, Xj1,
                                                   Wfij1, b1, attn1, elog1, fe_m, E);
    const long long EH1 = (long long)E * 3;
    seg_max<<<(unsigned)((EH1 + T - 1) / T), T, 0, stream>>>(elog1, dst, (unsigned int*)m1, E, 3);
    seg_expsum<<<(unsigned)((EH1 + T - 1) / T), T, 0, stream>>>(elog1, dst, (unsigned int*)m1,
                                                                s1, ex1, E, 3);
    // one wave per edge -> 8 edges per 256-thread block
    agg1<<<(E + 7) / 8, T, 0, stream>>>(src, dst, Hn1, ex1, s1, hmean, E);

    // ---- layer 2 node projections (WMMA fp32) ----
    dim3 g2((N + 63) / 64, 1);
    gemm_f32_wmma<<<g2, gblk, 0, stream>>>(hmean, Wni2, Xi2, N, 8, 32);
    gemm_f32_wmma<<<g2, gblk, 0, stream>>>(hmean, Wnj2, Xj2, N, 8, 32);
    gemm_f32_wmma<<<g2, gblk, 0, stream>>>(hmean, Wnode2, Hn2, N, 5, 32);

    // ---- layer 2 edge pipeline ----
    edge2_fused<<<(E + T - 1) / T, T, 0, stream>>>(fe_m, src, dst, Xi2, Xj2,
                                                   Wfij2, b2, attn2, elog2, E);
    seg_max<<<(E + T - 1) / T, T, 0, stream>>>(elog2, dst, (unsigned int*)m2, E, 1);
    seg_expsum<<<(E + T - 1) / T, T, 0, stream>>>(elog2, dst, (unsigned int*)m2, s2, ex2, E, 1);
    agg2<<<(E + T - 1) / T, T, 0, stream>>>(src, dst, Hn2, ex2, s2, h2, E);

    // ---- predictor ----
    predict<<<(E + T - 1) / T, T, 0, stream>>>(src, dst, h2, Wp, bp, (float*)d_out, E);
}